// BilateralFilter_OutofMemory_84129819394379
// MI455X (gfx1250) — compile-verified
//
#include <hip/hip_runtime.h>

// Bilateral 3D filter, 5x5x5, replicate padding, sigma_density=0.8, spatial sigma=1.1.
// out = sum_k g_k * exp2(-1.1271055*d_k^2) * p_k / sum_k g_k * exp2(-1.1271055*d_k^2)
// (both normalizations in the reference cancel in the ratio).
//
// Layout: one wave (32 lanes) handles 16 consecutive voxels along W; lanes m and
// m+16 redundantly compute voxel m's full (num, den). A single
// V_WMMA_F32_16X16X4_F32 with an all-ones B reduces across lanes:
// D[m, n] = num(lane m) + num(lane m+16) = 2*num_m  (factor 2 cancels in num/den).
//
// Perf notes vs previous round:
//  - seg/z/y/c/zz/yy/rowb forced wave-uniform via readfirstlane -> SGPR address
//    bases, per-tap address is one v_add_nc_u32 + saddr global_load (scale_offset).
//  - Taps staged one 5x5 slab (25 loads) at a time into registers so loads are
//    batched (1 wait per slab instead of 1 wait per load).

typedef __attribute__((ext_vector_type(2))) float v2f;
typedef __attribute__((ext_vector_type(8))) float v8f;

#define N48 48

__device__ __forceinline__ float wexp(float dd) {
#if __has_builtin(__builtin_amdgcn_exp2f)
  // exp(-dd/1.28) == exp2(-dd * 0.78125 * log2(e))
  return __builtin_amdgcn_exp2f(-1.12710550069f * dd);
#else
  return __expf(-0.78125f * dd);
#endif
}

__global__ __launch_bounds__(256) void bilateral3d_wmma(const float* __restrict__ X,
                                                        float* __restrict__ O) {
  const int lane = threadIdx.x & 31;
  const int wave = threadIdx.x >> 5;
  // Wave-uniform segment id, made provably uniform so coords land in SGPRs.
  const int seg  = __builtin_amdgcn_readfirstlane(blockIdx.x * 8 + wave);
  const int base = seg << 4;                // first flat voxel of the 16-voxel segment

  int t = base;
  const int x0 = t % N48; t /= N48;         // x0 in {0,16,32}: segment never crosses a row
  const int y  = t % N48; t /= N48;
  const int z  = t % N48;
  const int c  = t / N48;
  const int m  = lane & 15;
  const int x  = x0 + m;                    // per-lane

  const float* Xc = X + c * (N48 * N48 * N48);

  // Replicate-pad clamps. zz/yy are wave-uniform (SGPR); xx is per-lane.
  int zz[5], yy[5], xx[5];
#pragma unroll
  for (int i = 0; i < 5; ++i) {
    int a = z + i - 2; zz[i] = a < 0 ? 0 : (a > 47 ? 47 : a);
    a = y + i - 2;     yy[i] = a < 0 ? 0 : (a > 47 ? 47 : a);
    a = x + i - 2;     xx[i] = a < 0 ? 0 : (a > 47 ? 47 : a);
  }

  const float center = Xc[(z * N48 + y) * N48 + x];

  // Unnormalized separable spatial Gaussian, sigma = 0.3*((5-1)/2 - 1) + 0.8 = 1.1
  const float G[5] = {0.19149590f, 0.66151512f, 1.0f, 0.66151512f, 0.19149590f};

  float num = 0.0f, den = 0.0f;
#pragma unroll
  for (int kd = 0; kd < 5; ++kd) {
    // Stage the whole 5x5 slab first: 25 batched loads, single wait.
    float pv[25];
#pragma unroll
    for (int kh = 0; kh < 5; ++kh) {
      const int rowb = (zz[kd] * N48 + yy[kh]) * N48;   // SGPR
#pragma unroll
      for (int kw = 0; kw < 5; ++kw)
        pv[kh * 5 + kw] = Xc[rowb + xx[kw]];
    }
#pragma unroll
    for (int kh = 0; kh < 5; ++kh) {
      const float gdh = G[kd] * G[kh];
#pragma unroll
      for (int kw = 0; kw < 5; ++kw) {
        const float p = pv[kh * 5 + kw];
        const float d = p - center;
        const float e = wexp(d * d);
        const float g = gdh * G[kw];
        num = fmaf(g * e, p, num);
        den = fmaf(g, e, den);
      }
    }
  }

#if __has_builtin(__builtin_amdgcn_wmma_f32_16x16x4_f32)
  // A (16x4 f32): lanes 0-15 give K=0,1 (v0,v1); lanes 16-31 give K=2,3.
  // Put (num|den) in slot 0, zero in slot 1; B = all-ones -> D[m,*] = 2*sum_m.
  v2f an; an.x = num; an.y = 0.0f;
  v2f ad; ad.x = den; ad.y = 0.0f;
  v2f bo; bo.x = 1.0f; bo.y = 1.0f;
  v8f cz = {0.0f, 0.0f, 0.0f, 0.0f, 0.0f, 0.0f, 0.0f, 0.0f};
  v8f cn = __builtin_amdgcn_wmma_f32_16x16x4_f32(
      false, an, false, bo, (short)0, cz, false, false);
  v8f cd = __builtin_amdgcn_wmma_f32_16x16x4_f32(
      false, ad, false, bo, (short)0, cz, false, false);

  // D layout: VGPR j, lanes 0-15 -> row j (voxels 0-7); lanes 16-31 -> row j+8.
  const int j = lane & 7;
  float rn = cn[0], rd = cd[0];
  rn = (j == 1) ? cn[1] : rn;  rd = (j == 1) ? cd[1] : rd;
  rn = (j == 2) ? cn[2] : rn;  rd = (j == 2) ? cd[2] : rd;
  rn = (j == 3) ? cn[3] : rn;  rd = (j == 3) ? cd[3] : rd;
  rn = (j == 4) ? cn[4] : rn;  rd = (j == 4) ? cd[4] : rd;
  rn = (j == 5) ? cn[5] : rn;  rd = (j == 5) ? cd[5] : rd;
  rn = (j == 6) ? cn[6] : rn;  rd = (j == 6) ? cd[6] : rd;
  rn = (j == 7) ? cn[7] : rn;  rd = (j == 7) ? cd[7] : rd;

  const bool active = (lane & 8) == 0;                 // lanes 0-7 and 16-23
  const int vox = base + ((lane >> 4) << 3) + j;       // voxels 0-7 / 8-15
  if (active) O[vox] = rn / rd;                        // 2*num / 2*den == num/den
#else
  if (lane < 16) O[base + m] = num / den;
#endif
}

extern "C" void kernel_launch(void* const* d_in, const int* in_sizes, int n_in,
                              void* d_out, int out_size, void* d_ws, size_t ws_size,
                              hipStream_t stream) {
  (void)n_in; (void)d_ws; (void)ws_size; (void)out_size;
  const float* X = (const float*)d_in[0];
  float* O = (float*)d_out;
  const int nvox = in_sizes[0];           // 331776 = 3*48^3
  const int blocks = nvox / 128;          // 16 voxels/wave * 8 waves/block = 2592
  bilateral3d_wmma<<<dim3(blocks), dim3(256), 0, stream>>>(X, O);
}